// LocalSingularityStrength_56186762166869
// MI455X (gfx1250) — compile-verified
//
#include <hip/hip_runtime.h>
#include <stdint.h>

// ---------------------------------------------------------------------------
// LocalSingularityStrength on MI455X (gfx1250)
//
// Pass 1: per-sample min/max (block reduce + order-invariant encoded atomics)
// Pass 2: fused rescale + multi-scale box filter (doubling) + log-slope + BN,
//         rows streamed into LDS with the CDNA5 Tensor Data Mover
//         (tensor_load_to_lds, TENSORcnt) double-buffered per workgroup.
// ---------------------------------------------------------------------------

namespace {
constexpr int Bb = 16, Hh = 224, Wd = 224, Cc = 32;
constexpr int TW = 32;             // output columns per workgroup
constexpr int SH = 56;             // output rows per workgroup
constexpr int XCOLS = TW + 15;     // 47 staged input columns (halo -7..+8)
constexpr int H2C = 46;            // box2 columns w0-7 .. w0+38
constexpr int TSTEPS = SH + 15;    // 71 streamed input rows per strip
constexpr float EPSK = 1e-7f;
constexpr float BNEPS = 1e-3f;
}

typedef __attribute__((ext_vector_type(4))) unsigned tdm_u4;
typedef __attribute__((ext_vector_type(8))) int      tdm_i8;
typedef __attribute__((ext_vector_type(4))) int      tdm_i4;

// One contiguous 1-D tile: nelem fp32 -> LDS. D# per CDNA5 ISA 8.3/8.4:
// group0 = {count=1, lds_addr, global_addr lo, global_addr hi | type=2};
// group1 (loop-invariant, hoisted by caller) carries data_size=4B,
// tensor_dim0 = tile_dim0 = nelem, tensor_dim1 = tile_dim1 = 1.
// 6-arg builtin on this toolchain:
//   (u32x4 g0, i32x8 g1, i32x4 g2, i32x4 g3, i32x8 extra, i32 cpol)
__device__ __forceinline__ void tdm_issue(const float* gsrc, unsigned lds_byte,
                                          tdm_i8 g1) {
  unsigned long long ga = (unsigned long long)(uintptr_t)gsrc;
  tdm_u4 g0;
  g0[0] = 1u;                                 // count=1, user descriptor
  g0[1] = lds_byte;                           // LDS destination (bytes)
  g0[2] = (unsigned)ga;                       // global_addr[31:0]
  g0[3] = (unsigned)(ga >> 32) | (2u << 30);  // global_addr[56:32] | type=2
  tdm_i4 z4 = (tdm_i4)0;                      // groups 2/3 unused (<= 2D)
  tdm_i8 z8 = (tdm_i8)0;                      // extra group unused
  __builtin_amdgcn_tensor_load_to_lds(g0, g1, z4, z4, z8, 0);
}

__device__ __forceinline__ tdm_i8 tdm_make_g1(int nelem) {
  tdm_i8 g1;
  g1[0] = (2 << 16);                                    // data_size = 4 bytes
  g1[1] = (int)((unsigned)(nelem & 0xFFFF) << 16);      // tensor_dim0[15:0]
  g1[2] = (int)(((unsigned)nelem >> 16) | (1u << 16));  // dim0 hi | tensor_dim1=1
  g1[3] = (int)((unsigned)(nelem & 0xFFFF) << 16);      // tile_dim0 = nelem
  g1[4] = 1;                                            // tile_dim1 = 1
  g1[5] = nelem;                                        // tensor_dim0_stride lo
  g1[6] = 0;
  g1[7] = 0;
  return g1;
}

// Order-preserving float<->uint encode for deterministic atomic min/max.
__device__ __forceinline__ unsigned fenc(float f) {
  unsigned u = __float_as_uint(f);
  return (u & 0x80000000u) ? ~u : (u | 0x80000000u);
}
__device__ __forceinline__ float fdec(unsigned u) {
  return __uint_as_float((u & 0x80000000u) ? (u & 0x7FFFFFFFu) : ~u);
}

// ---------------------------------------------------------------------------
// Pass 1: per-sample min/max
// ---------------------------------------------------------------------------
__global__ void lss_ws_init(unsigned* ws) {
  int i = threadIdx.x;
  if (i < 16) ws[i] = 0u;                // encoded-max accumulators
  else if (i < 32) ws[i] = 0xFFFFFFFFu;  // encoded-min accumulators
}

__global__ __launch_bounds__(256) void lss_minmax(const float* __restrict__ x,
                                                  unsigned* __restrict__ ws) {
  __shared__ float smx[256];
  __shared__ float smn[256];
  const int b = blockIdx.y;
  const size_t N = (size_t)Hh * Wd * Cc;
  const size_t base = (size_t)b * N;
  const size_t stride = (size_t)gridDim.x * 256;
  float mx = -3.402823466e38f, mn = 3.402823466e38f;
  for (size_t i = (size_t)blockIdx.x * 256 + threadIdx.x; i < N; i += stride) {
    float v = x[base + i];
    mx = fmaxf(mx, v);
    mn = fminf(mn, v);
  }
  smx[threadIdx.x] = mx;
  smn[threadIdx.x] = mn;
  __syncthreads();
  for (int s = 128; s > 0; s >>= 1) {
    if (threadIdx.x < s) {
      smx[threadIdx.x] = fmaxf(smx[threadIdx.x], smx[threadIdx.x + s]);
      smn[threadIdx.x] = fminf(smn[threadIdx.x], smn[threadIdx.x + s]);
    }
    __syncthreads();
  }
  if (threadIdx.x == 0) {
    atomicMax(&ws[b], fenc(smx[0]));
    atomicMin(&ws[16 + b], fenc(smn[0]));
  }
}

__global__ void lss_finalize(unsigned* ws) {
  int b = threadIdx.x;
  if (b < 16) {
    float mx = fdec(ws[b]);
    float mn = fdec(ws[16 + b]);
    float scale = 1.0f / (mx - mn + EPSK);
    float* sf = (float*)(ws + 32);
    sf[2 * b + 0] = scale;       // x_scaled = x*scale + offset
    sf[2 * b + 1] = -mn * scale;
  }
}

// ---------------------------------------------------------------------------
// Pass 2: fused multi-scale box filter + log-slope + BatchNorm
//
// Horizontal: one LDS relay (box2); box4/8/16 are sums of box2 at odd offsets
//   H4[w]  = H2[w-1]+H2[w+1]
//   H8[w]  = H2[w-3]+H2[w-1]+H2[w+1]+H2[w+3]
//   H16[w] = sum of H2[w-7..w+7 odd]
// Vertical: per-thread register rings (mod 16) + sliding-window sums.
// Only 2 __syncthreads() per streamed row.
// ---------------------------------------------------------------------------
__global__ __launch_bounds__(1024, 1) void lss_main_kernel(
    const float* __restrict__ x,
    const float* __restrict__ gamma, const float* __restrict__ beta,
    const float* __restrict__ mmean, const float* __restrict__ mvar,
    const float* __restrict__ sstats,   // [2*B] scale, offset
    float* __restrict__ out) {
  __shared__ float xbuf[2][XCOLS * Cc];  // TDM double-buffered row staging
  __shared__ float h2buf[H2C * Cc];      // horizontal box2 relay

  const int tid = threadIdx.x;
  const int w0 = blockIdx.x * TW;
  const int h0 = blockIdx.y * SH;
  const int b  = blockIdx.z;
  const int c  = tid & (Cc - 1);   // lane = channel -> coalesced 128B/wave
  const int wl = tid >> 5;         // owned output column within tile

  const float scl = sstats[2 * b + 0];
  const float off = sstats[2 * b + 1];
  const float gn = gamma[c] * rsqrtf(mvar[c] + BNEPS);
  const float bt = fmaf(-mmean[c], gn, beta[c]);

  // Clamped contiguous global span for this tile's rows (never reads OOB).
  const int gstart = (w0 - 7 < 0) ? 0 : (w0 - 7);
  const int gend   = (w0 + 40 > Wd) ? Wd : (w0 + 40);
  const int nelem  = (gend - gstart) * Cc;
  const unsigned ldsshift = (unsigned)((gstart - (w0 - 7)) * Cc * 4);

  const unsigned xlds[2] = {
      (unsigned)(uintptr_t)&xbuf[0][0] + ldsshift,
      (unsigned)(uintptr_t)&xbuf[1][0] + ldsshift};

  const tdm_i8 g1 = tdm_make_g1(nelem);     // loop-invariant D# group1
  constexpr size_t ROWSTRIDE = (size_t)Wd * Cc;

  // Vertical register ring buffers (mod 16) + sliding-window sums.
  float r2[16], r4[16], r8[16], r16[16];
#pragma unroll
  for (int i = 0; i < 16; ++i) { r2[i] = 0.f; r4[i] = 0.f; r8[i] = 0.f; r16[i] = 0.f; }
  float s2 = 0.f, s4 = 0.f, s8 = 0.f, s16 = 0.f;

  const int rS = h0 - 7;  // first streamed input row
  // Running pointer to the row currently being consumed (row r).
  const float* grow = x + (((size_t)b * Hh + rS) * Wd + gstart) * Cc;

  // Prime the pipeline: TDM-load row rS into xbuf[0].
  if (tid < 32) {
    if (rS >= 0 && rS < Hh) tdm_issue(grow, xlds[0], g1);
    __builtin_amdgcn_s_wait_tensorcnt(0);
  }
  __syncthreads();

  for (int t0 = 0; t0 < 80; t0 += 16) {
#pragma unroll
    for (int u = 0; u < 16; ++u) {
      const int t = t0 + u;
      if (t < TSTEPS) {
        const int r = rS + t;   // input row being consumed this step
        const int p = t & 1;

        // 1) Kick off next row's TDM copy into the other staging buffer.
        const int rn = r + 1;
        if (tid < 32 && (t + 1) < TSTEPS && rn >= 0 && rn < Hh) {
          tdm_issue(grow + ROWSTRIDE, xlds[p ^ 1], g1);
        }

        const bool rowv = (r >= 0) && (r < Hh);
        const float* xb = xbuf[p];

        // 2) Horizontal box2 with fused min-max rescale + zero padding.
        for (int j = tid; j < H2C * Cc; j += 1024) {
          const int col = j >> 5, ch = j & 31;
          const int wg0 = w0 - 7 + col;
          float a0 = (rowv && wg0 >= 0 && wg0 < Wd)
                         ? fmaf(xb[col * Cc + ch], scl, off) : 0.f;
          float a1 = (rowv && (wg0 + 1) >= 0 && (wg0 + 1) < Wd)
                         ? fmaf(xb[(col + 1) * Cc + ch], scl, off) : 0.f;
          h2buf[j] = a0 + a1;
        }
        __syncthreads();

        // 3) All horizontal sums for this thread's column from box2 alone.
        const float b0  = h2buf[(wl + 0)  * Cc + c];
        const float b2  = h2buf[(wl + 2)  * Cc + c];
        const float b4  = h2buf[(wl + 4)  * Cc + c];
        const float b6  = h2buf[(wl + 6)  * Cc + c];
        const float b7  = h2buf[(wl + 7)  * Cc + c];
        const float b8  = h2buf[(wl + 8)  * Cc + c];
        const float b10 = h2buf[(wl + 10) * Cc + c];
        const float b12 = h2buf[(wl + 12) * Cc + c];
        const float b14 = h2buf[(wl + 14) * Cc + c];
        const float p46   = b4 + b6;
        const float p810  = b8 + b10;
        const float p02   = b0 + b2;
        const float p1214 = b12 + b14;
        const float h2v  = b7;
        const float h4v  = b6 + b8;
        const float h8v  = p46 + p810;
        const float h16v = (p02 + p46) + (p810 + p1214);

        // 4) Vertical sliding windows (all aligned to output row i = r-8):
        //    M2: rows i..i+1, M4: i-1..i+2, M8: i-3..i+4, M16: i-7..i+8.
        const float old16 = r16[u];
        r2[u] = h2v; r4[u] = h4v; r8[u] = h8v; r16[u] = h16v;
        s16 += h16v - old16;                           // +H16[r]  -H16[r-16]
        s2  += r2[(u + 9)  & 15] - r2[(u + 7)  & 15];  // +H2[r-7] -H2[r-9]
        s4  += r4[(u + 10) & 15] - r4[(u + 6)  & 15];  // +H4[r-6] -H4[r-10]
        s8  += r8[(u + 12) & 15] - r8[(u + 4)  & 15];  // +H8[r-4] -H8[r-12]

        // 5) Emit output row i = r - 8 (valid once rings are warm).
        if (t >= 15) {
          const int i = r - 8;
          const float l2  = __log2f(s2  + EPSK);
          const float l4  = __log2f(s4  + EPSK);
          const float l8  = __log2f(s8  + EPSK);
          const float l16 = __log2f(s16 + EPSK);
          // alpha = sum(c_s * ln(m_s)) / (5 ln^2 2) * ln 2  ==  log2 form / 5
          const float alpha = (1.5f * (l16 - l2) + 0.5f * (l8 - l4)) * 0.2f;
          out[(((size_t)b * Hh + i) * Wd + (w0 + wl)) * Cc + c] =
              fmaf(alpha, gn, bt);
        }

        grow += ROWSTRIDE;

        // 6) Close the step: next row's TDM tile must have landed, and all
        //    waves must be done with h2buf before it is rewritten.
        if (tid < 32) __builtin_amdgcn_s_wait_tensorcnt(0);
        __syncthreads();
      }
    }
  }
}

// ---------------------------------------------------------------------------
extern "C" void kernel_launch(void* const* d_in, const int* in_sizes, int n_in,
                              void* d_out, int out_size, void* d_ws,
                              size_t ws_size, hipStream_t stream) {
  const float* x     = (const float*)d_in[0];
  const float* gamma = (const float*)d_in[1];
  const float* beta  = (const float*)d_in[2];
  const float* mmean = (const float*)d_in[3];
  const float* mvar  = (const float*)d_in[4];
  float* out = (float*)d_out;
  unsigned* ws = (unsigned*)d_ws;  // [0,16): enc-max, [16,32): enc-min,
                                   // [32,64): per-sample {scale, offset} (fp32)

  lss_ws_init<<<1, 64, 0, stream>>>(ws);
  lss_minmax<<<dim3(64, Bb), 256, 0, stream>>>(x, ws);
  lss_finalize<<<1, 16, 0, stream>>>(ws);
  lss_main_kernel<<<dim3(Wd / TW, Hh / SH, Bb), 1024, 0, stream>>>(
      x, gamma, beta, mmean, mvar, (const float*)(ws + 32), out);
}